// FMFMNeuron_42262478192716
// MI455X (gfx1250) — compile-verified
//
#include <hip/hip_runtime.h>
#include <stdint.h>

// ---------------- problem constants (from reference) ----------------
#define T_STEPS 4096
#define BATCH   4096
#define COLS    32                 // columns handled per workgroup (one wave32 lane each)
#define CHUNK   128                // timesteps per TDM tile
#define NCHUNK  (T_STEPS / CHUNK)  // 32
#define STAGES  3                  // LDS ring depth (pipeline depth)
#define IN_ROW  (COLS * 2)         // 64 floats = 256B per timestep row (s0,s1 interleaved)
// LDS: 3*128*64*4 = 96KB input ring + 3*128*32*4 = 48KB output ring = 144KB

typedef unsigned int u32x4 __attribute__((ext_vector_type(4)));
typedef int          i32x4 __attribute__((ext_vector_type(4)));
typedef int          i32x8 __attribute__((ext_vector_type(8)));

#if defined(__HIP_DEVICE_COMPILE__) && \
    __has_builtin(__builtin_amdgcn_tensor_load_to_lds) && \
    __has_builtin(__builtin_amdgcn_tensor_store_from_lds) && \
    __has_builtin(__builtin_amdgcn_s_wait_tensorcnt)
#define USE_TDM 1
#else
#define USE_TDM 0
#endif

#if USE_TDM
// Build a 2-D Tensor DMA Descriptor (D# groups 0 and 1) per CDNA5 ISA ch.8.
// data_size = 4 bytes; tile rows of tile_d0 elements, tile_d1 rows,
// row stride stride0 elements in global memory; LDS rows packed contiguously.
__device__ __forceinline__ void tdm_desc(u32x4& g0, i32x8& g1,
                                         uint64_t gaddr, uint32_t ldsaddr,
                                         uint32_t tensor_d0, uint32_t tensor_d1,
                                         uint64_t stride0,
                                         uint32_t tile_d0, uint32_t tile_d1)
{
  // group0: [1:0] count=1 | [63:32] lds_addr | [120:64] global_addr | [127:126] type=2
  g0[0] = 1u;
  g0[1] = ldsaddr;
  g0[2] = (uint32_t)gaddr;
  g0[3] = ((uint32_t)(gaddr >> 32) & 0x01FFFFFFu) | (2u << 30);
  // group1: wg_mask=0 | data_size=2(4B) | dims/strides per bit layout
  g1[0] = (int)(2u << 16);                                              // data_size=4B
  g1[1] = (int)((tensor_d0 & 0xFFFFu) << 16);                           // dim0[15:0]
  g1[2] = (int)((tensor_d0 >> 16) | ((tensor_d1 & 0xFFFFu) << 16));     // dim0[31:16]|dim1[15:0]
  g1[3] = (int)((tensor_d1 >> 16) | (tile_d0 << 16));                   // dim1[31:16]|tile0
  g1[4] = (int)(tile_d1 & 0xFFFFu);                                     // tile1 (tile2=0)
  g1[5] = (int)(uint32_t)stride0;                                       // stride0[31:0]
  g1[6] = (int)((uint32_t)(stride0 >> 32) & 0xFFFFu);                   // stride0[47:32]
  g1[7] = 0;
}
#endif

extern "C" __global__ void __launch_bounds__(COLS)
lif_scan_tdm(const float* __restrict__ spikes,  // [T, B, 2]
             const float* __restrict__ W,       // [1, 2]
             float* __restrict__ out)           // [T, B, 1]
{
  extern __shared__ float smem[];
  const int   wg   = blockIdx.x;
  const int   lane = threadIdx.x;
  const float w1   = W[0];
  const float w2   = W[1];

  float mem = 0.0f, prevspk = 0.0f;  // reset_t == spk_{t-1} (threshold = 1.0)

#if USE_TDM
  float* in_buf  = smem;
  float* out_buf = smem + STAGES * CHUNK * IN_ROW;

  const uint32_t lds_in  = (uint32_t)(uintptr_t)(void*)in_buf;
  const uint32_t lds_out = (uint32_t)(uintptr_t)(void*)out_buf;
  const uint64_t in_base  = (uint64_t)(uintptr_t)spikes + (uint64_t)wg * (COLS * 2 * sizeof(float));
  const uint64_t out_base = (uint64_t)(uintptr_t)out    + (uint64_t)wg * (COLS * sizeof(float));
  const i32x4 gz4 = {0, 0, 0, 0};
  const i32x8 gz8 = {0, 0, 0, 0, 0, 0, 0, 0};

  // ---- prologue: fill the 3-deep input ring ----
  for (int c = 0; c < STAGES; ++c) {
    u32x4 g0; i32x8 g1;
    tdm_desc(g0, g1,
             in_base + (uint64_t)c * CHUNK * (BATCH * 2 * 4),
             lds_in + (uint32_t)((c % STAGES) * CHUNK * IN_ROW * 4),
             BATCH * 2, T_STEPS, BATCH * 2, IN_ROW, CHUNK);
    __builtin_amdgcn_tensor_load_to_lds(g0, g1, gz4, gz4, gz8, 0);
  }

  for (int c = 0; c < NCHUNK; ++c) {
    // TDM ops complete in order per wave: issued = 3+2c; wait until L_c done
    // (this also guarantees S_{c-3}, which precedes L_c, has drained its out-stage).
    if (c == 0)      __builtin_amdgcn_s_wait_tensorcnt(2);
    else if (c == 1) __builtin_amdgcn_s_wait_tensorcnt(3);
    else             __builtin_amdgcn_s_wait_tensorcnt(4);
    __asm__ volatile("" ::: "memory");

    const float* ib = in_buf  + (c % STAGES) * (CHUNK * IN_ROW);
    float*       ob = out_buf + (c % STAGES) * (CHUNK * COLS);

#pragma unroll 4
    for (int t = 0; t < CHUNK; ++t) {
      const float s0  = ib[t * IN_ROW + lane * 2 + 0];
      const float s1  = ib[t * IN_ROW + lane * 2 + 1];
      const float cur = __builtin_fmaf(s1, w2, s0 * w1);
      mem = __builtin_fmaf(0.95f, mem, cur - prevspk);       // beta*mem + cur - reset*1.0
      const float spk = (mem > 1.0f) ? 1.0f : 0.0f;
      prevspk = spk;
      ob[t * COLS + lane] = spk;
    }

    // All LDS writes (and reads) must land before TDM touches these stages.
    __asm__ volatile("s_wait_dscnt 0" ::: "memory");

    {  // drain output chunk c via TDM
      u32x4 g0; i32x8 g1;
      tdm_desc(g0, g1,
               out_base + (uint64_t)c * CHUNK * (BATCH * 4),
               lds_out + (uint32_t)((c % STAGES) * CHUNK * COLS * 4),
               BATCH, T_STEPS, BATCH, COLS, CHUNK);
      __builtin_amdgcn_tensor_store_from_lds(g0, g1, gz4, gz4, gz8, 0);
    }
    if (c + STAGES < NCHUNK) {  // refill the input stage we just consumed
      const int n = c + STAGES;
      u32x4 g0; i32x8 g1;
      tdm_desc(g0, g1,
               in_base + (uint64_t)n * CHUNK * (BATCH * 2 * 4),
               lds_in + (uint32_t)((n % STAGES) * CHUNK * IN_ROW * 4),
               BATCH * 2, T_STEPS, BATCH * 2, IN_ROW, CHUNK);
      __builtin_amdgcn_tensor_load_to_lds(g0, g1, gz4, gz4, gz8, 0);
    }
  }
  __builtin_amdgcn_s_wait_tensorcnt(0);
#else
  // Fallback (no TDM builtins): direct streaming scan, identical mapping.
  const int b = wg * COLS + lane;
  for (int t = 0; t < T_STEPS; ++t) {
    const float s0  = spikes[((size_t)t * BATCH + b) * 2 + 0];
    const float s1  = spikes[((size_t)t * BATCH + b) * 2 + 1];
    const float cur = __builtin_fmaf(s1, w2, s0 * w1);
    mem = __builtin_fmaf(0.95f, mem, cur - prevspk);
    const float spk = (mem > 1.0f) ? 1.0f : 0.0f;
    prevspk = spk;
    out[(size_t)t * BATCH + b] = spk;
  }
#endif
}

extern "C" void kernel_launch(void* const* d_in, const int* in_sizes, int n_in,
                              void* d_out, int out_size, void* d_ws, size_t ws_size,
                              hipStream_t stream) {
  (void)in_sizes; (void)n_in; (void)out_size; (void)d_ws; (void)ws_size;
  const float* spikes = (const float*)d_in[0];
  const float* W      = (const float*)d_in[1];
  float*       out    = (float*)d_out;

  const size_t smem_bytes = (size_t)STAGES * CHUNK * IN_ROW * sizeof(float)   // 96 KB
                          + (size_t)STAGES * CHUNK * COLS   * sizeof(float);  // 48 KB
  dim3 grid(BATCH / COLS);  // 128 workgroups
  dim3 block(COLS);         // 1 wave32 each
  hipLaunchKernelGGL(lif_scan_tdm, grid, block, smem_bytes, stream, spikes, W, out);
}